// SOM2dLayer_23029614641678
// MI455X (gfx1250) — compile-verified
//
#include <hip/hip_runtime.h>
#include <hip/hip_bf16.h>

// SOM BMU search for MI455X (gfx1250, wave32).
//
// Roofline: 2*B*N*D = 17.2 GFLOP fp32 GEMM vs 16 MB of matrix data (L2-resident
// in the 192MB L2) -> compute-bound. Use the exact-fp32 matrix instruction
// V_WMMA_F32_16X16X4_F32 so the argmin is bit-trustworthy (bf16 WMMA would
// risk BMU flips on near-ties). Distance via ||w||^2 - 2 x.w (row-constant
// ||x||^2 added only for the quantization error).
//
// v3: software-pipelined inner loop. v2's assembly showed every WMMA pair
// stalling on a just-issued global_load_b64 (s_wait_loadcnt 0x0 right before
// each v_wmma). Now K is processed in chunks of 8 k-steps with register
// double-buffering: chunk c+1's 24 loads are issued before chunk c's 16 WMMAs,
// so the WMMA burst overlaps the L1/L2 latency of the next chunk's loads.

typedef __attribute__((ext_vector_type(2))) float v2f;
typedef __attribute__((ext_vector_type(8))) float v8f;

#define BATCH 4096
#define DIM   512
#define NNEUR 4096   // 64*64
#define WGRID 64
#define KCH   8      // k-steps (of 4) per software-pipeline chunk
#define NCH   ((DIM / 4) / KCH)   // 16 chunks

// --- kernel 1: wsq[n] = sum_k W[n][k]^2 ------------------------------------
__global__ __launch_bounds__(128)
void som_wsq_kernel(const float* __restrict__ Wg, float* __restrict__ wsq) {
    __shared__ float sred[4];
    const int n = blockIdx.x;
    const int t = threadIdx.x;
    const float* wr = Wg + (size_t)n * DIM;
    float s = 0.f;
#pragma unroll
    for (int j = 0; j < DIM / 128; ++j) {
        float v = wr[t + 128 * j];
        s = fmaf(v, v, s);
    }
    // wave32 reduction
#pragma unroll
    for (int m = 1; m < 32; m <<= 1) s += __shfl_xor(s, m);
    if ((t & 31) == 0) sred[t >> 5] = s;
    __syncthreads();
    if (t == 0) wsq[n] = sred[0] + sred[1] + sred[2] + sred[3];
}

// --- kernel 2: fused GEMM (WMMA f32 16x16x4) + argmin + qerror -------------
// grid = B/16 workgroups, 256 threads = 8 wave32s per WG.
// Each WG owns a 16-row X tile (LDS); each wave sweeps PAIRS of 16-col W tiles.
__global__ __launch_bounds__(256)
void som_bmu_kernel(const float* __restrict__ X,
                    const float* __restrict__ Wg,
                    const float* __restrict__ wsq,
                    float* __restrict__ out) {
    __shared__ float Xs[16 * DIM];     // 32 KB X tile, reused 8 waves x 16 pair-tiles
    __shared__ float xsq_s[16];
    __shared__ float redv[8][16];
    __shared__ int   redi[8][16];

    const int tid   = threadIdx.x;
    const int wv    = tid >> 5;        // wave id 0..7
    const int lane  = tid & 31;
    const int half  = lane >> 4;       // 0: K{0,1}/M rows 0..7 ; 1: K{2,3}/M rows 8..15
    const int l16   = lane & 15;
    const int mbase = blockIdx.x * 16;

    // Stage X tile into LDS and fuse per-row ||x||^2. 16 threads per row.
    {
        const int row = tid >> 4;      // 0..15
        const int c0  = tid & 15;
        const float* xr = X + (size_t)(mbase + row) * DIM;
        float* xs = Xs + row * DIM;
        float ss = 0.f;
#pragma unroll
        for (int j = 0; j < DIM / 16; ++j) {
            float v = xr[c0 + 16 * j];
            xs[c0 + 16 * j] = v;
            ss = fmaf(v, v, ss);
        }
        // 16 row-mates are a contiguous half-wave: xor-reduce within it
        ss += __shfl_xor(ss, 1);
        ss += __shfl_xor(ss, 2);
        ss += __shfl_xor(ss, 4);
        ss += __shfl_xor(ss, 8);
        if (c0 == 0) xsq_s[row] = ss;
    }
    __syncthreads();

    // Running (min t, argmin n) per accumulator slot r.
    // Slot r of this lane corresponds to row m = mbase + half*8 + r.
    float run_v[8];
    int   run_i[8];
#pragma unroll
    for (int r = 0; r < 8; ++r) { run_v[r] = 3.4e38f; run_i[r] = 0x7fffffff; }

    const int koff = 2 * half;                    // A/B fragment K sub-offset
    const float* xrow = Xs + l16 * DIM + koff;    // A: row m=l16 of X tile

    // 128 pair-tiles of 32 columns; wave wv takes pairs wv, wv+8, ...
    for (int p = wv; p < NNEUR / 32; p += 8) {
        const int nbase = p * 32;
        const int n0    = nbase + l16;            // lane's column in tile 0
        const int n1    = n0 + 16;                // lane's column in tile 1
        const float* wrow0 = Wg + (size_t)n0 * DIM + koff;
        const float* wrow1 = Wg + (size_t)n1 * DIM + koff;

        v8f acc0 = {};                            // two independent C chains
        v8f acc1 = {};

        // ---- software pipeline over K: chunks of KCH k-steps ----
        v2f aC[KCH], b0C[KCH], b1C[KCH];          // current chunk fragments
#pragma unroll
        for (int j = 0; j < KCH; ++j) {           // prologue: chunk 0 loads
            aC[j]  = *(const v2f*)(xrow  + 4 * j);
            b0C[j] = *(const v2f*)(wrow0 + 4 * j);
            b1C[j] = *(const v2f*)(wrow1 + 4 * j);
        }
#pragma unroll 2
        for (int c = 0; c < NCH; ++c) {
            v2f aN[KCH], b0N[KCH], b1N[KCH];
            if (c + 1 < NCH) {                    // issue chunk c+1 loads first
                const int kb = (c + 1) * KCH;
#pragma unroll
                for (int j = 0; j < KCH; ++j) {
                    aN[j]  = *(const v2f*)(xrow  + 4 * (kb + j));
                    b0N[j] = *(const v2f*)(wrow0 + 4 * (kb + j));
                    b1N[j] = *(const v2f*)(wrow1 + 4 * (kb + j));
                }
            }
            // 16 WMMAs on chunk c overlap chunk c+1's outstanding loads
#pragma unroll
            for (int j = 0; j < KCH; ++j) {
                acc0 = __builtin_amdgcn_wmma_f32_16x16x4_f32(
                    false, aC[j], false, b0C[j], (short)0, acc0, false, false);
                acc1 = __builtin_amdgcn_wmma_f32_16x16x4_f32(
                    false, aC[j], false, b1C[j], (short)0, acc1, false, false);
            }
#pragma unroll
            for (int j = 0; j < KCH; ++j) {       // rotate buffers
                aC[j] = aN[j]; b0C[j] = b0N[j]; b1C[j] = b1N[j];
            }
        }

        const float wq0 = wsq[n0];
        const float wq1 = wsq[n1];
#pragma unroll
        for (int r = 0; r < 8; ++r) {
            float t0 = fmaf(-2.f, acc0[r], wq0);  // ||w||^2 - 2 x.w
            if (t0 < run_v[r] || (t0 == run_v[r] && n0 < run_i[r])) {
                run_v[r] = t0; run_i[r] = n0;
            }
            float t1 = fmaf(-2.f, acc1[r], wq1);
            if (t1 < run_v[r] || (t1 == run_v[r] && n1 < run_i[r])) {
                run_v[r] = t1; run_i[r] = n1;
            }
        }
    }

    // Cross-lane argmin within each 16-lane half (columns l16 = 0..15).
#pragma unroll
    for (int r = 0; r < 8; ++r) {
        float v = run_v[r]; int i = run_i[r];
#pragma unroll
        for (int m = 1; m <= 8; m <<= 1) {
            float ov = __shfl_xor(v, m);
            int   oi = __shfl_xor(i, m);
            if (ov < v || (ov == v && oi < i)) { v = ov; i = oi; }
        }
        run_v[r] = v; run_i[r] = i;
    }
    if (l16 == 0) {
#pragma unroll
        for (int r = 0; r < 8; ++r) {
            redv[wv][half * 8 + r] = run_v[r];
            redi[wv][half * 8 + r] = run_i[r];
        }
    }
    __syncthreads();

    // Combine the 8 waves; 16 threads finalize one row each.
    if (tid < 16) {
        const int row = tid;
        float v = redv[0][row]; int i = redi[0][row];
#pragma unroll
        for (int k = 1; k < 8; ++k) {
            float ov = redv[k][row]; int oi = redi[k][row];
            if (ov < v || (ov == v && oi < i)) { v = ov; i = oi; }
        }
        const int gm = mbase + row;
        float sq = fmaxf(xsq_s[row] + v, 0.f);
        out[2 * gm + 0]     = (float)(i >> 6);    // bmu_y = n / 64
        out[2 * gm + 1]     = (float)(i & 63);    // bmu_x = n % 64
        out[2 * BATCH + gm] = sqrtf(sq);          // quantization error
    }
}

extern "C" void kernel_launch(void* const* d_in, const int* in_sizes, int n_in,
                              void* d_out, int out_size, void* d_ws, size_t ws_size,
                              hipStream_t stream) {
    const float* X  = (const float*)d_in[0];   // (4096, 512) fp32
    const float* Wg = (const float*)d_in[1];   // (64, 64, 512) fp32 -> (4096, 512)
    float* out = (float*)d_out;                // 8192 index floats + 4096 qerrors
    float* wsq = (float*)d_ws;                 // 4096 floats of scratch

    som_wsq_kernel<<<NNEUR, 128, 0, stream>>>(Wg, wsq);
    som_bmu_kernel<<<BATCH / 16, 256, 0, stream>>>(X, Wg, wsq, out);
}